// Fine_Change_Moment3_28656021799222
// MI455X (gfx1250) — compile-verified
//
#include <hip/hip_runtime.h>

typedef float v2f __attribute__((ext_vector_type(2)));
typedef float v8f __attribute__((ext_vector_type(8)));

#define Bsz 16
#define Ssz 512
#define Tsz 16
#define Psz 4
#define Gsz 128                    // Ssz / Psz
#define NPATCH (Bsz * Gsz * Gsz)   // 262144

// ---------------------------------------------------------------------------
// Phase 1: per-patch CAM channel 0 via chained V_WMMA_F32_16X16X4_F32.
// One full wave (32 lanes) per patch, EXEC all-ones through the WMMAs.
//   A (16x4 f32): lanes 0-15 hold M=t, v0=K, v1=K+1; lanes 16-31 hold K+2/K+3.
//   B (4x16 f32): broadcast weight column -> every D column holds cam0[t].
//   D (16x16 f32): lane j: VGPR v = cam0[v] (j<16) / cam0[v+8] (j>=16).
// Normalize (x - min)/(max - min) over t, store top in (b,gy,gx,t) layout.
// ---------------------------------------------------------------------------
__global__ __launch_bounds__(256) void cam_phase1(const float* __restrict__ in,
                                                  const float* __restrict__ w3,
                                                  float* __restrict__ top)
{
    const int lane  = threadIdx.x & 31;
    const int wave  = threadIdx.x >> 5;
    const int patch = blockIdx.x * 8 + wave;      // exact grid, no guard

    const int b  = patch >> 14;                   // / (128*128)
    const int r  = patch & 16383;
    const int gy = r >> 7;
    const int gx = r & 127;

    const int tm = lane & 15;                     // M index (channel t)
    const int hi = lane >> 4;                     // lane group
    const int c0 = hi ? 2 : 0;                    // K offset within a patch row

    const size_t inbase = (((size_t)b * Ssz + (size_t)gy * Psz) * Ssz + (size_t)gx * Psz) * Tsz;
    const size_t wbase  = (size_t)patch * (Psz * Psz * 6);   // fc_weight3[n, k, 0] at +k*6

    __builtin_prefetch(&w3[wbase], 0, 0);         // global_prefetch_b8 on strided weight col

    v8f d = {0.f, 0.f, 0.f, 0.f, 0.f, 0.f, 0.f, 0.f};

#pragma unroll
    for (int s = 0; s < 4; ++s) {
        // K block = [4s, 4s+3]; this lane supplies K = 4s+c0 and 4s+c0+1.
        const int k0 = 4 * s + c0;
        v2f a, bb;
        // fm[t, k] = input[b, gy*4 + k/4, gx*4 + k%4, t]; k/4 == s, k%4 == c0
        const size_t ra = inbase + ((size_t)s * Ssz + (size_t)c0) * Tsz + (size_t)tm;
        a.x  = in[ra];            // 16 lanes -> 64B coalesced
        a.y  = in[ra + Tsz];
        bb.x = w3[wbase + (size_t)k0 * 6];        // broadcast across N (same addr / group)
        bb.y = w3[wbase + (size_t)(k0 + 1) * 6];
        d = __builtin_amdgcn_wmma_f32_16x16x4_f32(false, a, false, bb,
                                                  (short)0, d, false, false);
    }

    // min/max over this lane's 8 M values, then combine the two lane groups.
    float mn = d[0], mx = d[0];
#pragma unroll
    for (int v = 1; v < 8; ++v) { mn = fminf(mn, d[v]); mx = fmaxf(mx, d[v]); }
    mn = fminf(mn, __shfl_xor(mn, 16));
    mx = fmaxf(mx, __shfl_xor(mx, 16));

    // Dynamic select d[lane & 7] via cndmask tree (no scratch).
    const int vs = lane & 7;
    float s0 = (vs & 1) ? d[1] : d[0];
    float s1 = (vs & 1) ? d[3] : d[2];
    float s2 = (vs & 1) ? d[5] : d[4];
    float s3 = (vs & 1) ? d[7] : d[6];
    float p0 = (vs & 2) ? s1 : s0;
    float p1 = (vs & 2) ? s3 : s2;
    float mine  = (vs & 4) ? p1 : p0;             // grp0: cam0[vs]; grp1: cam0[vs+8]
    float other = __shfl_xor(mine, 16);           // value from the other lane group

    const int  out_t = lane & 15;
    const bool lo    = out_t < 8;
    const bool grp0  = lane < 16;
    float val  = lo ? (grp0 ? mine : other) : (grp0 ? other : mine);
    float norm = (val - mn) / (mx - mn);

    if (lane < 16)                                 // 64B store per patch
        top[(size_t)patch * Tsz + out_t] = norm;
}

// ---------------------------------------------------------------------------
// Phase 2: bilinear upsample 128 -> 512, align_corners=True (scale 127/511).
// One thread = one float4 (4 channels of one output pixel); top stays in L2.
// out[((b*S + o)*S + p)*T + t] layout == reference (-1, T) flatten.
// ---------------------------------------------------------------------------
__global__ __launch_bounds__(256) void upsample_phase2(const float4* __restrict__ top4,
                                                       float4* __restrict__ out4)
{
    const int idx = blockIdx.x * 256 + threadIdx.x;   // 16,777,216 threads
    const int tq  = idx & 3;                          // float4 group within T=16
    const int pix = idx >> 2;
    const int p   = pix & 511;
    const int o   = (pix >> 9) & 511;
    const int b   = pix >> 18;

    const float sc = 127.0f / 511.0f;
    float co = (float)o * sc; int i0 = (int)co; if (i0 > 126) i0 = 126; float wi = co - (float)i0;
    float cp = (float)p * sc; int j0 = (int)cp; if (j0 > 126) j0 = 126; float wj = cp - (float)j0;

    const size_t base = (((size_t)b * Gsz + (size_t)i0) * Gsz + (size_t)j0) * 4 + (size_t)tq;
    float4 t00 = top4[base];
    float4 t01 = top4[base + 4];                  // j0+1
    float4 t10 = top4[base + Gsz * 4];            // i0+1
    float4 t11 = top4[base + Gsz * 4 + 4];

    const float w00 = (1.f - wi) * (1.f - wj);
    const float w01 = (1.f - wi) * wj;
    const float w10 = wi * (1.f - wj);
    const float w11 = wi * wj;

    float4 r;
    r.x = w00 * t00.x + w01 * t01.x + w10 * t10.x + w11 * t11.x;
    r.y = w00 * t00.y + w01 * t01.y + w10 * t10.y + w11 * t11.y;
    r.z = w00 * t00.z + w01 * t01.z + w10 * t10.z + w11 * t11.z;
    r.w = w00 * t00.w + w01 * t01.w + w10 * t10.w + w11 * t11.w;

    out4[idx] = r;                                // fully coalesced 128-bit stores
}

extern "C" void kernel_launch(void* const* d_in, const int* in_sizes, int n_in,
                              void* d_out, int out_size, void* d_ws, size_t ws_size,
                              hipStream_t stream) {
    const float* in = (const float*)d_in[0];   // (16,512,512,16) f32
    const float* w3 = (const float*)d_in[1];   // (262144,16,6)  f32
    float* top = (float*)d_ws;                 // 262144*16 f32 = 16 MB scratch

    cam_phase1<<<NPATCH / 8, 256, 0, stream>>>(in, w3, top);

    const int n2 = (Bsz * Ssz * Ssz * Tsz) / 4;       // 16,777,216 float4 threads
    upsample_phase2<<<n2 / 256, 256, 0, stream>>>((const float4*)top, (float4*)d_out);
}